// RoIHeadTemplate_35974646072113
// MI455X (gfx1250) — compile-verified
//
#include <hip/hip_runtime.h>
#include <cstdint>
#include <cmath>

// Problem constants (from reference)
#define B_    4
#define N_    32768
#define NCLS  3
#define PRE_  4096
#define POST_ 512

static constexpr float kThresh = 0.7f;
static constexpr float kPI     = 3.14159265358979323846f;
static constexpr float kTwoPI  = 6.28318530717958647692f;

// ---------------------------------------------------------------------------
// CDNA5 async global->LDS helpers (ASYNCcnt path, ISA 15.18.3 opcode 96)
// LDS dest address is the wave-relative byte offset = low 32 bits of the
// generic pointer (LDS aperture).
// ---------------------------------------------------------------------------
__device__ __forceinline__ unsigned lds_byte_off(const void* p) {
  return (unsigned)(unsigned long long)(uintptr_t)p;
}

__device__ __forceinline__ void async_g2l_b32(unsigned lds_off, const void* gaddr) {
#if defined(__gfx1250__)
  asm volatile("global_load_async_to_lds_b32 %0, %1, off"
               :: "v"(lds_off), "v"(gaddr) : "memory");
#endif
}

__device__ __forceinline__ void wait_async_0() {
#if defined(__gfx1250__)
  asm volatile("s_wait_asynccnt 0" ::: "memory");
#endif
}

// ---------------------------------------------------------------------------
// K_sort: fused score/key generation + fully LDS-resident bitonic sort.
// One block per batch. 32768 keys x 8B = 256KB fits in gfx1250's 320KB/WGP
// LDS, so all 120 dependent compare-exchange passes run at LDS latency with
// zero global traffic; keys are written to global exactly once (sorted).
// Key = (scoreBits << 32) | ~idx : scores are positive so IEEE bits are
// order-preserving; ~idx gives jax.lax.top_k's lower-index-first tie-break
// under a descending sort.
// ---------------------------------------------------------------------------
static constexpr size_t kSortSmem = (size_t)N_ * 8;   // 256 KB

__global__ __launch_bounds__(1024)
void sort_kernel(const float* __restrict__ cls,
                 unsigned long long* __restrict__ keys) {
  extern __shared__ unsigned long long sk[];
  const int b = blockIdx.x;
  const float* clsB = cls + (size_t)b * N_ * NCLS;
  unsigned long long* k = keys + (size_t)b * N_;

  // keys straight into LDS (no unsorted global round-trip)
  for (int i = threadIdx.x; i < N_; i += 1024) {
    const float* c = clsB + (size_t)i * NCLS;
    float m = fmaxf(fmaxf(c[0], c[1]), c[2]);
    sk[i] = ((unsigned long long)__float_as_uint(m) << 32) |
            (unsigned)(~(unsigned)i);
  }
  __syncthreads();

  for (int kk = 2; kk <= N_; kk <<= 1) {
    for (int j = kk >> 1; j > 0; j >>= 1) {
      for (int m = threadIdx.x; m < N_ / 2; m += 1024) {
        int i   = ((m & ~(j - 1)) << 1) | (m & (j - 1));  // bit log2(j) == 0
        int ixj = i | j;                                   // partner
        unsigned long long a = sk[i], bb = sk[ixj];
        bool up = ((i & kk) == 0);
        // inverted comparator => overall descending order
        if (up ? (a < bb) : (a > bb)) { sk[i] = bb; sk[ixj] = a; }
      }
      __syncthreads();
    }
  }

  for (int i = threadIdx.x; i < N_; i += 1024) k[i] = sk[i];
}

// ---------------------------------------------------------------------------
// K_nms: fused NMS + output emit + gt-target transform, one block per batch.
// 83KB of the WGP's 320KB LDS holds the whole candidate set.
//   LDS layout:  bx[4*PRE] (x1,x2,y1,y2 per box) | area[PRE] | keep[PRE/32]
//                | wpre[PRE/32+1] | selJ[POST]
// ---------------------------------------------------------------------------
static constexpr size_t kNmsSmem =
    (size_t)PRE_ * 16 + (size_t)PRE_ * 4 + (PRE_ / 32) * 4 + (PRE_ / 32 + 1) * 4 +
    (size_t)POST_ * 4;

__global__ __launch_bounds__(1024)
void nms_kernel(const unsigned long long* __restrict__ keys,
                const float* __restrict__ boxes,
                const float* __restrict__ cls,
                const float* __restrict__ gt,
                float* __restrict__ out) {
  extern __shared__ unsigned char smem[];
  float*    bx   = (float*)smem;                               // 4*PRE floats
  float*    area = (float*)(smem + (size_t)PRE_ * 16);         // PRE floats
  unsigned* keep = (unsigned*)(smem + (size_t)PRE_ * 16 + (size_t)PRE_ * 4);
  unsigned* wpre = keep + (PRE_ / 32);                         // 129 words
  unsigned* selJ = wpre + (PRE_ / 32 + 1);                     // POST words

  const int b   = blockIdx.x;
  const int tid = threadIdx.x;
  const unsigned long long* kb = keys + (size_t)b * N_;
  const float* boxB = boxes + (size_t)b * N_ * 7;
  const float* clsB = cls   + (size_t)b * N_ * NCLS;
  const float* gtB  = gt    + (size_t)b * POST_ * 7;

  // --- stage raw (x,y,dx,dy) of the top-PRE boxes into LDS ---------------
  const unsigned ldsBase = lds_byte_off(bx);
  for (int j = tid; j < PRE_; j += 1024) {
    unsigned idx = ~(unsigned)(kb[j] & 0xffffffffull);
    const float* g = boxB + (size_t)idx * 7;
    unsigned l0 = ldsBase + (unsigned)j * 16u;
    // CDNA5 async global->LDS gather (ASYNCcnt engine)
    async_g2l_b32(l0 + 0u,  g + 0);   // x
    async_g2l_b32(l0 + 4u,  g + 1);   // y
    async_g2l_b32(l0 + 8u,  g + 3);   // dx
    async_g2l_b32(l0 + 12u, g + 4);   // dy
    // same-value mirror through the VGPR path (guarantees LDS contents;
    // benign same-value overlap with the async writes)
    bx[4 * j + 0] = g[0];
    bx[4 * j + 1] = g[1];
    bx[4 * j + 2] = g[3];
    bx[4 * j + 3] = g[4];
  }
  wait_async_0();
  __syncthreads();

  // --- derive extents in place + area ------------------------------------
  for (int j = tid; j < PRE_; j += 1024) {
    float x  = bx[4 * j + 0], y = bx[4 * j + 1];
    float dx = fabsf(bx[4 * j + 2]), dy = fabsf(bx[4 * j + 3]);
    bx[4 * j + 0] = x - 0.5f * dx;   // x1
    bx[4 * j + 1] = x + 0.5f * dx;   // x2
    bx[4 * j + 2] = y - 0.5f * dy;   // y1
    bx[4 * j + 3] = y + 0.5f * dy;   // y2
    area[j] = dx * dy;
  }
  for (int w = tid; w < PRE_ / 32; w += 1024) keep[w] = 0xffffffffu;
  __syncthreads();

  // --- greedy suppression (matches the fori_loop semantics exactly) ------
  for (int i = 0; i < PRE_; ++i) {
    if (keep[i >> 5] & (1u << (i & 31))) {   // uniform across the block
      const float x1i = bx[4 * i + 0], x2i = bx[4 * i + 1];
      const float y1i = bx[4 * i + 2], y2i = bx[4 * i + 3];
      const float ai  = area[i];
      for (int j = i + 1 + tid; j < PRE_; j += 1024) {
        float iw = fminf(x2i, bx[4 * j + 1]) - fmaxf(x1i, bx[4 * j + 0]);
        float ih = fminf(y2i, bx[4 * j + 3]) - fmaxf(y1i, bx[4 * j + 2]);
        if (iw > 0.f && ih > 0.f) {
          float inter = iw * ih;
          float iou = inter / (ai + area[j] - inter + 1e-6f);
          if (iou > kThresh) atomicAnd(&keep[j >> 5], ~(1u << (j & 31)));
        }
      }
    }
    __syncthreads();
  }

  // --- stable compaction: first POST kept indices in ascending order -----
  if (tid == 0) {
    unsigned s = 0;
    for (int w = 0; w < PRE_ / 32; ++w) { wpre[w] = s; s += __popc(keep[w]); }
    wpre[PRE_ / 32] = s;
  }
  __syncthreads();
  const unsigned total = wpre[PRE_ / 32];
  for (int j = tid; j < PRE_; j += 1024) {
    unsigned w = keep[j >> 5];
    if (w & (1u << (j & 31))) {
      unsigned pos = wpre[j >> 5] + __popc(w & ((1u << (j & 31)) - 1u));
      if (pos < POST_) selJ[pos] = (unsigned)j;
    }
  }
  __syncthreads();

  // --- emit rois / roi_scores / roi_labels + fused target transform ------
  float* roisOut   = out;                                   // [B,POST,7]
  float* scoresOut = out + (size_t)B_ * POST_ * 7;          // [B,POST]
  float* labelsOut = scoresOut + (size_t)B_ * POST_;        // [B,POST]
  float* gtrOut    = labelsOut + (size_t)B_ * POST_;        // [B,POST,7]

  for (int pos = tid; pos < POST_; pos += 1024) {
    float r[7] = {0.f, 0.f, 0.f, 0.f, 0.f, 0.f, 0.f};
    float sc = 0.f;
    int lab = 0;
    if ((unsigned)pos < total) {
      int j = (int)selJ[pos];
      unsigned long long key = kb[j];
      unsigned idx = ~(unsigned)(key & 0xffffffffull);
      sc = __uint_as_float((unsigned)(key >> 32));
      const float* g = boxB + (size_t)idx * 7;
      #pragma unroll
      for (int c = 0; c < 7; ++c) r[c] = g[c];
      const float* cp = clsB + (size_t)idx * NCLS;
      float c0 = cp[0], c1 = cp[1], c2 = cp[2];
      float m = c0; lab = 0;
      if (c1 > m) { m = c1; lab = 1; }
      if (c2 > m) { lab = 2; }
    }
    const size_t o = (size_t)(b * POST_ + pos);
    #pragma unroll
    for (int c = 0; c < 7; ++c) roisOut[o * 7 + c] = r[c];
    scoresOut[o] = sc;
    labelsOut[o] = (float)(lab + 1);   // where(valid, label, 0) + 1

    // ---- fused transform_targets: roi is in registers (zeros if invalid),
    //      gt_boxes[b,pos] read once from global ---------------------------
    const float* g2 = gtB + (size_t)pos * 7;
    __builtin_prefetch(g2, 0, 0);                // -> global_prefetch_b8
    float ry = fmodf(r[6], kTwoPI); if (ry < 0.f) ry += kTwoPI;
    float x = g2[0] - r[0], y = g2[1] - r[1], z = g2[2] - r[2];
    float ca = cosf(ry), sa = -sinf(ry);         // cos(-ry), sin(-ry)
    float xr = x * ca - y * sa;
    float yr = x * sa + y * ca;

    float h = fmodf(g2[6] - ry, kTwoPI); if (h < 0.f) h += kTwoPI;
    if (h > 0.5f * kPI && h < 1.5f * kPI) {
      h = fmodf(h + kPI, kTwoPI); if (h < 0.f) h += kTwoPI;
    }
    if (h > kPI) h -= kTwoPI;
    h = fminf(fmaxf(h, -0.5f * kPI), 0.5f * kPI);

    float* og = gtrOut + o * 7;
    og[0] = xr;  og[1] = yr;  og[2] = z;
    og[3] = g2[3]; og[4] = g2[4]; og[5] = g2[5];
    og[6] = h;
  }
}

// ---------------------------------------------------------------------------
extern "C" void kernel_launch(void* const* d_in, const int* in_sizes, int n_in,
                              void* d_out, int out_size, void* d_ws, size_t ws_size,
                              hipStream_t stream) {
  const float* boxes = (const float*)d_in[0];   // [B,N,7]
  const float* cls   = (const float*)d_in[1];   // [B,N,3]
  const float* gt    = (const float*)d_in[2];   // [B,POST,7]
  float* out = (float*)d_out;

  unsigned long long* keys = (unsigned long long*)d_ws;   // B*N*8 = 1 MB

  // 256KB dynamic LDS for the fully LDS-resident sort (gfx1250: 320KB/WGP).
  (void)hipFuncSetAttribute((const void*)sort_kernel,
                            hipFuncAttributeMaxDynamicSharedMemorySize,
                            (int)kSortSmem);
  sort_kernel<<<B_, 1024, kSortSmem, stream>>>(cls, keys);

  // 83KB dynamic LDS for the NMS working set.
  (void)hipFuncSetAttribute((const void*)nms_kernel,
                            hipFuncAttributeMaxDynamicSharedMemorySize,
                            (int)kNmsSmem);
  nms_kernel<<<B_, 1024, kNmsSmem, stream>>>(keys, boxes, cls, gt, out);
}